// Net_91268055040039
// MI455X (gfx1250) — compile-verified
//
#include <hip/hip_runtime.h>

typedef __attribute__((ext_vector_type(2))) float v2f;
typedef __attribute__((ext_vector_type(8))) float v8f;
typedef __attribute__((ext_vector_type(4))) unsigned int v4u;
typedef __attribute__((ext_vector_type(8))) int v8i;
typedef __attribute__((ext_vector_type(4))) int v4i;

#define NB      32                 // batches
#define K_TOT   4000000            // T*P = 400*10000
#define KCHUNK  256                // k per block iteration (15625 chunks exactly)
#define NCHUNK  (K_TOT / KCHUNK)   // 15625
#define ROWPAD  4                  // 4-DWORD pad -> row stride 260 (bank stride 4, 16B aligned)
#define ROWSTR  (KCHUNK + ROWPAD)  // 260 floats per LDS row

// Seed out[b] = fc1_b every launch (atomic accumulation must start clean per replay).
__global__ void init_out_kernel(float* __restrict__ out, const float* __restrict__ fc1_b) {
    if (threadIdx.x < NB) out[threadIdx.x] = fc1_b[0];
}

// TDM: DMA one 32-row x 256-col f32 tile of x into LDS, with hardware row
// padding of 4 DWORDs every 256 DWORDs (reproduces ROWSTR=260 layout).
// D# bit layout per CDNA5 ISA ch.8 (groups 0/1; 2D tensor -> groups 2/3 zero).
__device__ __forceinline__ void tdm_load_x_tile(const float* gsrc, float* lds_dst) {
    const unsigned lds_off = (unsigned)(size_t)(void*)lds_dst;      // addr[31:0] = LDS byte offset
    const unsigned long long ga = (unsigned long long)(size_t)gsrc; // 57-bit global byte address

    v4u g0;
    g0[0] = 1u;                                  // count=1 (valid user descriptor)
    g0[1] = lds_off;                             // lds_addr [63:32]
    g0[2] = (unsigned)ga;                        // global_addr [95:64]
    g0[3] = (unsigned)(ga >> 32) | (2u << 30);   // global_addr [120:96], type=2 ("image")

    v8i g1;
    g1[0] = (int)((2u << 16)      // data_size = 2 -> 4 bytes
                | (1u << 20)      // pad_enable
                | (7u << 22)      // pad_interval: 256 DWORDs between pads
                | (3u << 25));    // pad_amount: 4 DWORDs
    g1[1] = (int)((K_TOT & 0xFFFFu) << 16);                      // tensor_dim0[15:0]
    g1[2] = (int)(((K_TOT >> 16) & 0xFFFFu) | ((unsigned)NB << 16)); // dim0[31:16] | dim1[15:0]
    g1[3] = (int)((unsigned)KCHUNK << 16);                       // dim1 hi=0 | tile_dim0=256
    g1[4] = NB;                                                  // tile_dim1=32, tile_dim2=0
    g1[5] = (int)K_TOT;                                          // tensor_dim0_stride lo32
    g1[6] = 0;                                                   // dim0_stride hi | dim1_stride lo
    g1[7] = 0;

    v4i z4 = {};
    v8i z8 = {};
    __builtin_amdgcn_tensor_load_to_lds(g0, g1, z4, z4, z8, 0);
}

// out[b] += sum_k x[b,k] * |W[k]| * fc1_w[k]
// TDM streams x into double-buffered padded LDS tiles while waves run
// V_WMMA_F32_16X16X4_F32 on the other buffer (f32-exact accumulation).
__global__ __launch_bounds__(256, 2)
void dotv_wmma_kernel(const float* __restrict__ x,
                      const float* __restrict__ W,
                      const float* __restrict__ fc,
                      float* __restrict__ out) {
    __shared__ float xtile[2][NB * ROWSTR];      // 2 x ~33 KB padded x tiles
    __shared__ float vbuf[2][KCHUNK];            // 2 x 1 KB fused v = |W|*fc1_w

    const int tid  = threadIdx.x;
    const int wave = tid >> 5;                   // 0..7
    const int lane = tid & 31;
    const int m    = lane & 15;                  // A-matrix row within batch group
    const int hi   = lane >> 4;                  // lane half: k = vgpr + 2*hi
    const int grp  = wave & 1;                   // batch rows [grp*16, grp*16+16)
    const int sec  = wave >> 1;                  // k-section 0..3 (64 k each)

    v8f acc = {};

    // ---- Prologue: stage first chunk into buffer 0 ----
    int chunk = blockIdx.x;                      // always < NCHUNK (grid clamped)
    if (wave == 0) tdm_load_x_tile(x + (size_t)chunk * KCHUNK, &xtile[0][0]);
    {
        const int gk = chunk * KCHUNK + tid;
        vbuf[0][tid] = __builtin_fabsf(W[gk]) * fc[gk];
    }
    if (wave == 0) __builtin_amdgcn_s_wait_tensorcnt(0);
    __syncthreads();

    int cur = 0;
    for (; chunk < NCHUNK; chunk += gridDim.x) {
        const int nxtc = chunk + gridDim.x;
        const bool hasNext = nxtc < NCHUNK;

        // Kick off next chunk: TDM for x, plain loads for W/fc (consumed after WMMA).
        float wv = 0.0f, fv = 0.0f;
        if (hasNext) {
            if (wave == 0) tdm_load_x_tile(x + (size_t)nxtc * KCHUNK, &xtile[cur ^ 1][0]);
            const int gk = nxtc * KCHUNK + tid;
            wv = W[gk];
            fv = fc[gk];
        }

        // ---- WMMA on current buffer: 16 batches x 64 k per wave ----
        const float* xrow = &xtile[cur][(grp * 16 + m) * ROWSTR];
        const float* vrow = &vbuf[cur][0];
        #pragma unroll
        for (int s = 0; s < 16; ++s) {
            const int c = sec * 64 + s * 4 + 2 * hi;   // k = (c-base) + vgpr index
            v2f a = *(const v2f*)&xrow[c];             // A: x[m, k..k+1]
            v2f b = *(const v2f*)&vrow[c];             // B: v[k..k+1] (all N columns equal)
            acc = __builtin_amdgcn_wmma_f32_16x16x4_f32(
                      false, a, false, b, (short)0, acc, false, false);
        }

        if (hasNext) vbuf[cur ^ 1][tid] = __builtin_fabsf(wv) * fv;
        if (wave == 0) __builtin_amdgcn_s_wait_tensorcnt(0);
        __syncthreads();
        cur ^= 1;
    }

    // Column N=0 of D: lane 0 holds M=0..7 in acc[0..7], lane 16 holds M=8..15.
    if (m == 0) {
        const int mb = grp * 16 + hi * 8;
        #pragma unroll
        for (int r = 0; r < 8; ++r)
            atomicAdd(&out[mb + r], acc[r]);
    }
}

extern "C" void kernel_launch(void* const* d_in, const int* in_sizes, int n_in,
                              void* d_out, int out_size, void* d_ws, size_t ws_size,
                              hipStream_t stream) {
    const float* x   = (const float*)d_in[0];   // [32, 400, 10000] f32
    const float* W   = (const float*)d_in[1];   // [400, 10000] f32 (flat K)
    const float* fcw = (const float*)d_in[2];   // [1, 4000000] f32
    const float* fcb = (const float*)d_in[3];   // [1] f32
    float* out = (float*)d_out;                 // [32] f32

    hipLaunchKernelGGL(init_out_kernel, dim3(1), dim3(32), 0, stream, out, fcb);

    int nblk = 2048;
    if (nblk > NCHUNK) nblk = NCHUNK;
    hipLaunchKernelGGL(dotv_wmma_kernel, dim3(nblk), dim3(256), 0, stream,
                       x, W, fcw, out);
}